// SRSMLP_45028437131994
// MI455X (gfx1250) — compile-verified
//
#include <hip/hip_runtime.h>
#include <hip/hip_bf16.h>

typedef __attribute__((ext_vector_type(2))) float v2f;
typedef __attribute__((ext_vector_type(8))) float v8f;

// Kernel 1: start = clamp(sum(mask[B-1,:]) - x, 0, S - x), written to ws[0].
// Single block, shared-memory tree reduction (no atomics -> no ws pre-init needed).
__global__ void SRSMLP_start_kernel(const int* __restrict__ mask,
                                    const int* __restrict__ xp,
                                    int* __restrict__ ws, int B, int S) {
    __shared__ int sdata[256];
    const int tid = threadIdx.x;
    const long long base = (long long)(B - 1) * (long long)S;
    int sum = 0;
    for (int i = tid; i < S; i += blockDim.x) sum += mask[base + i];
    sdata[tid] = sum;
    __syncthreads();
    for (int s = 128; s > 0; s >>= 1) {
        if (tid < s) sdata[tid] += sdata[tid + s];
        __syncthreads();
    }
    if (tid == 0) {
        const int x = xp[0];
        int start = sdata[0] - x;           // dynamic_slice clamps:
        if (start < 0) start = 0;
        if (start > S - x) start = S - x;
        ws[0] = start;
    }
}

// Kernel 2: ht[b,d] = sum_{k<x} hidden[b, start+k, d] via V_WMMA_F32_16X16X4_F32.
// One wave32 per 16-output tile (one batch b, 16 consecutive d).
// A[m,k] = hidden window element (m = d-offset 0..15, k = 4 consecutive s rows),
// B = ones(4x16)  =>  D[m,n] = sum_k A[m,k] for all n; accumulate C over x/4 steps.
__global__ void SRSMLP_sumwin_wmma_kernel(const float* __restrict__ hidden,
                                          const int* __restrict__ start_p,
                                          const int* __restrict__ xp,
                                          float* __restrict__ out,
                                          int B, int S, int D) {
    const int wavesPerBlock = blockDim.x >> 5;
    const int wave = blockIdx.x * wavesPerBlock + (threadIdx.x >> 5);
    const int lane = threadIdx.x & 31;

    const int tilesPerBatch = D >> 4;                 // 64 for D=1024
    const int b  = wave / tilesPerBatch;
    if (b >= B) return;                               // wave-uniform: EXEC stays all-1s
    const int d0 = (wave % tilesPerBatch) << 4;

    const int start = start_p[0];
    const int nkb   = xp[0] >> 2;                     // x/4 WMMA accumulation steps

    const int m     = lane & 15;                      // A-matrix M index (d offset)
    const int kbase = (lane >> 4) << 1;               // lanes 0-15: K=0,1; lanes 16-31: K=2,3

    // per-lane base pointer: row (start+kbase) of batch b, column d0+m
    const float* __restrict__ src =
        hidden + ((long long)b * S + start + kbase) * (long long)D + d0 + m;

    v2f bmat = {1.0f, 1.0f};                          // B = ones (all layouts identical)
    v8f c = {};

    for (int kb = 0; kb < nkb; ++kb) {
        const float* p = src + (long long)(kb << 2) * D;
        v2f a;
        a.x = p[0];                                   // K = kbase   (coalesced 64B/half-wave)
        a.y = p[D];                                   // K = kbase+1
        c = __builtin_amdgcn_wmma_f32_16x16x4_f32(
                /*neg_a=*/false, a, /*neg_b=*/false, bmat,
                /*c_mod=*/(short)0, c, /*reuse_a=*/false, /*reuse_b=*/false);
    }

    // C/D layout: lane j(<16) VGPR r holds D[M=r, N=j]; lane 16+j holds D[M=r+8, N=j].
    // All N columns identical -> lanes 0-7 store m=0..7, lanes 16-23 store m=8..15.
    float cr[8];
#pragma unroll
    for (int i = 0; i < 8; ++i) cr[i] = c[i];
    const int half = lane >> 4;
    const int j    = lane & 15;
    if (j < 8) out[wave * 16 + half * 8 + j] = cr[j];
}

extern "C" void kernel_launch(void* const* d_in, const int* in_sizes, int n_in,
                              void* d_out, int out_size, void* d_ws, size_t ws_size,
                              hipStream_t stream) {
    const float* hidden = (const float*)d_in[0];
    const int*   mask   = (const int*)d_in[1];
    const int*   xp     = (const int*)d_in[2];
    float*       out    = (float*)d_out;
    int*         ws     = (int*)d_ws;

    const int hiddenN = in_sizes[0];                  // B*S*D
    const int maskN   = in_sizes[1];                  // B*S
    const int D = hiddenN / maskN;                    // 1024
    const int B = out_size / D;                       // 32
    const int S = maskN / B;                          // 2048

    SRSMLP_start_kernel<<<1, 256, 0, stream>>>(mask, xp, ws, B, S);

    const int waves   = (B * D) >> 4;                 // one wave per 16 outputs = 2048
    const int threads = 256;                          // 8 wave32s per block
    const int blocks  = (waves * 32 + threads - 1) / threads;
    SRSMLP_sumwin_wmma_kernel<<<blocks, threads, 0, stream>>>(hidden, ws, xp, out, B, S, D);
}